// CustomSSIM_16853451669928
// MI455X (gfx1250) — compile-verified
//
#include <hip/hip_runtime.h>
#include <hip/hip_bf16.h>

// ---------------------------------------------------------------------------
// SSIM on MI455X (gfx1250): separable Gaussian blurs expressed as banded
// matmuls on V_WMMA_F32_16X16X4_F32 (wave32, one 16x16 output tile per wave).
//
// Roofline: 100.7 MB mandatory reads / 23.3 TB/s ~= 4.3 us floor; the f32
// convolution FLOPs only stay off the critical path on the matrix pipe, so
// all 10 blur passes (5 fields x H,W) run as 16x16x4 f32 WMMAs against a
// constant band matrix built from the Gaussian taps.
// ---------------------------------------------------------------------------

typedef float v2f __attribute__((ext_vector_type(2)));
typedef float v8f __attribute__((ext_vector_type(8)));

#define WIN 11
#define TILE 16
#define NCHUNK 7            // K = 26 padded to 28 = 7 chunks of 4
#define IMG_H 512
#define IMG_W 512
#define OUT_H 502           // 512 - 10 (VALID)
#define OUT_W 502
#define TILES_X 32          // ceil(502/16)
#define TILES_Y 32
#define PLANES 48           // 16 batches * 3 channels
#define WAVES_PER_BLOCK 4
#define NFIELD 5
#define LDS_STRIDE 34       // even stride: b64-aligned col pairs, no bank conflicts
#define GPAD 48             // zero-padded window table

__global__ void ssim_init_kernel(float* out) { out[0] = 0.0f; }

__global__ __launch_bounds__(WAVES_PER_BLOCK * 32)
void ssim_wmma_kernel(const float* __restrict__ img1,
                      const float* __restrict__ img2,
                      const float* __restrict__ gwin,
                      float* __restrict__ out)
{
    __shared__ float ldsT[WAVES_PER_BLOCK][NFIELD][TILE * LDS_STRIDE];
    __shared__ float gpad[GPAD];

    const int tid  = threadIdx.x;
    const int lane = tid & 31;
    const int wave = tid >> 5;
    const int tile = blockIdx.x * WAVES_PER_BLOCK + wave;

    const int plane = tile >> 10;       // tile / (32*32)
    const int rem   = tile & 1023;
    const int ty    = rem >> 5;
    const int tx    = rem & 31;

    const int half  = lane >> 4;        // 16-lane half
    const int mn    = lane & 15;        // A-row / B-col / D-col index
    const int kbase = half * 2;         // K sub-offset within a K=4 chunk

    // Zero-padded Gaussian band table: gpad[16 + d] = g[d] for d in [0,11).
    if (tid < GPAD) {
        const int d = tid - 16;
        gpad[tid] = (d >= 0 && d < WIN) ? gwin[d] : 0.0f;
    }
    __syncthreads();

    // Band fragments. Per lane the H-pass A fragment and W-pass B fragment
    // coincide: value = g[(4c + kbase + j) - mn]  (zero outside the band).
    v2f wfrag[NCHUNK];
#pragma unroll
    for (int c = 0; c < NCHUNK; ++c) {
        wfrag[c][0] = gpad[16 + 4 * c + kbase + 0 - mn];
        wfrag[c][1] = gpad[16 + 4 * c + kbase + 1 - mn];
    }

    const float* p1 = img1 + (size_t)plane * (IMG_H * IMG_W);
    const float* p2 = img2 + (size_t)plane * (IMG_H * IMG_W);
    const int r0 = ty * TILE;
    const int c0 = tx * TILE;

    // ---- H pass: for each col group, load (a1,a2) once, feed all 5 fields --
#pragma unroll
    for (int grp = 0; grp < 2; ++grp) {
        v8f t[NFIELD];
#pragma unroll
        for (int f = 0; f < NFIELD; ++f)
            t[f] = (v8f){0.f, 0.f, 0.f, 0.f, 0.f, 0.f, 0.f, 0.f};

        int cc = c0 + 16 * grp + mn;
        if (cc > IMG_W - 1) cc = IMG_W - 1;

#pragma unroll
        for (int c = 0; c < NCHUNK; ++c) {
            float a1[2], a2[2];
#pragma unroll
            for (int j = 0; j < 2; ++j) {
                int rr = r0 + 4 * c + kbase + j;
                if (rr > IMG_H - 1) rr = IMG_H - 1;   // clamped halo
                const size_t idx = (size_t)rr * IMG_W + cc;
                a1[j] = p1[idx];
                a2[j] = p2[idx];
            }
            const v2f b0 = {a1[0],         a1[1]};
            const v2f b1 = {a2[0],         a2[1]};
            const v2f b2 = {a1[0] * a1[0], a1[1] * a1[1]};
            const v2f b3 = {a2[0] * a2[0], a2[1] * a2[1]};
            const v2f b4 = {a1[0] * a2[0], a1[1] * a2[1]};
            t[0] = __builtin_amdgcn_wmma_f32_16x16x4_f32(false, wfrag[c], false, b0, (short)0, t[0], false, false);
            t[1] = __builtin_amdgcn_wmma_f32_16x16x4_f32(false, wfrag[c], false, b1, (short)0, t[1], false, false);
            t[2] = __builtin_amdgcn_wmma_f32_16x16x4_f32(false, wfrag[c], false, b2, (short)0, t[2], false, false);
            t[3] = __builtin_amdgcn_wmma_f32_16x16x4_f32(false, wfrag[c], false, b3, (short)0, t[3], false, false);
            t[4] = __builtin_amdgcn_wmma_f32_16x16x4_f32(false, wfrag[c], false, b4, (short)0, t[4], false, false);
        }

        // D layout: lane holds (M = v + 8*half, N = mn) of this col group.
#pragma unroll
        for (int f = 0; f < NFIELD; ++f) {
#pragma unroll
            for (int v = 0; v < 8; ++v) {
                ldsT[wave][f][(v + 8 * half) * LDS_STRIDE + 16 * grp + mn] = t[f][v];
            }
        }
    }

    __syncthreads();  // T visible before transposed (column-pair) reads

    // ---- W pass: Out(16x16) = T(16x28) x Band^T(28x16), per field ----------
    v8f acc[NFIELD];
#pragma unroll
    for (int f = 0; f < NFIELD; ++f) {
        v8f r = {0.f, 0.f, 0.f, 0.f, 0.f, 0.f, 0.f, 0.f};
        const float* Tf = &ldsT[wave][f][0];
#pragma unroll
        for (int c = 0; c < NCHUNK; ++c) {
            // index = mn*34 + 4c + kbase is even -> 8B-aligned ds_load_b64
            const float2 av = *(const float2*)&Tf[mn * LDS_STRIDE + 4 * c + kbase];
            const v2f a = {av.x, av.y};
            r = __builtin_amdgcn_wmma_f32_16x16x4_f32(
                    false, a, false, wfrag[c], (short)0, r, false, false);
        }
        acc[f] = r;
    }

    // ---- SSIM map + masked partial sum (D layout: M = v + 8*half, N = mn) --
    const float c1 = 0.01f * 0.01f;
    const float c2 = 0.03f * 0.03f;
    float partial = 0.0f;
#pragma unroll
    for (int v = 0; v < 8; ++v) {
        const int m = v + 8 * half;
        const bool valid = ((r0 + m) < OUT_H) && ((c0 + mn) < OUT_W);
        const float mu1  = acc[0][v];
        const float mu2  = acc[1][v];
        const float m11  = acc[2][v];
        const float m22  = acc[3][v];
        const float m12  = acc[4][v];
        const float mu1s = mu1 * mu1;
        const float mu2s = mu2 * mu2;
        const float mu12 = mu1 * mu2;
        const float s1   = m11 - mu1s;
        const float s2   = m22 - mu2s;
        const float s12  = m12 - mu12;
        const float cs   = (2.0f * s12 + c2) / (s1 + s2 + c2);
        const float sm   = ((2.0f * mu12 + c1) / (mu1s + mu2s + c1)) * cs;
        partial += valid ? sm : 0.0f;
    }

    // wave32 reduction
#pragma unroll
    for (int off = 16; off > 0; off >>= 1)
        partial += __shfl_xor(partial, off, 32);

    if (lane == 0) {
        const float inv_total =
            1.0f / ((float)PLANES * (float)OUT_H * (float)OUT_W);
        atomicAdd(out, partial * inv_total);
    }
}

extern "C" void kernel_launch(void* const* d_in, const int* in_sizes, int n_in,
                              void* d_out, int out_size, void* d_ws, size_t ws_size,
                              hipStream_t stream)
{
    (void)in_sizes; (void)n_in; (void)out_size; (void)d_ws; (void)ws_size;
    const float* img1 = (const float*)d_in[0];
    const float* img2 = (const float*)d_in[1];
    const float* gwin = (const float*)d_in[2];
    float* out = (float*)d_out;

    ssim_init_kernel<<<1, 1, 0, stream>>>(out);

    const int total_tiles = PLANES * TILES_X * TILES_Y;          // 49152
    const int blocks = total_tiles / WAVES_PER_BLOCK;            // 12288
    ssim_wmma_kernel<<<blocks, WAVES_PER_BLOCK * 32, 0, stream>>>(
        img1, img2, gwin, out);
}